// LDRLoss_80187039416585
// MI455X (gfx1250) — compile-verified
//
#include <hip/hip_runtime.h>
#include <cmath>

// Problem geometry (from the reference)
#define TLEN      (1 << 20)      // samples per row
#define NROWS     16             // batch
#define CHUNK     1024           // per-wave scan chunk
#define NCHUNK    (TLEN / CHUNK) // 1024
#define SHIFT     65047          // jnp.roll(-SHIFT) -> e(t) pairs with index t+SHIFT mod T
#define SHIFT_CH  (SHIFT / CHUNK)   // 63
#define SHIFT_OFF (SHIFT % CHUNK)   // 535

typedef __attribute__((ext_vector_type(2))) float v2f;
typedef __attribute__((ext_vector_type(8))) float v8f;
typedef __attribute__((ext_vector_type(4))) int   v4i;

// ---------------------------------------------------------------------------
// gfx1250 async global->LDS staging (ASYNCcnt) if the toolchain exposes it;
// otherwise fall back to global_load_b128 + ds_store_b128 (DScnt).
// Probe-confirmed: the builtin exists here and takes (v4i*, v4i*, imm, imm).
// ---------------------------------------------------------------------------
#if defined(__has_builtin)
#  if __has_builtin(__builtin_amdgcn_global_load_async_to_lds_b128)
#    define USE_ASYNC_LDS 1
#  endif
#endif
#ifndef USE_ASYNC_LDS
#  define USE_ASYNC_LDS 0
#endif

// Wave-local LDS synchronization. All LDS regions here are private to one
// wave, and same-wave DS ops are in-order (ISA 7.3), so a dscnt wait plus a
// compiler memory barrier replaces block-wide s_barrier entirely.
__device__ __forceinline__ void wave_wait_ds() {
  asm volatile("s_wait_dscnt 0x0" ::: "memory");
}
__device__ __forceinline__ void wave_wait_tile_published() {
#if USE_ASYNC_LDS
  asm volatile("s_wait_asynccnt 0x0" ::: "memory");
#else
  asm volatile("s_wait_dscnt 0x0" ::: "memory");
#endif
}

// ---------------------------------------------------------------------------
// Per-coefficient per-lane constants for the WMMA-based tile scan.
// Recurrence: y[t] = a*y[t-1] + c*x[t],  a = 1-c (constant per coefficient).
// A 256-sample tile is 16 segments (columns) x 16 steps (rows):
//   D = L * Bmat,  L[i][j] = a^(i-j) (i>=j, else 0),  Bmat[i][j] = c*x[16j+i]
// done as 4 chained V_WMMA_F32_16X16X4_F32 (K=16 as 4 x K=4).
// ---------------------------------------------------------------------------
__device__ __forceinline__ float ipow_f(float a, int d) {
  float p = 1.0f;
  for (int i = 0; i < d; ++i) p *= a;
  return p;
}

struct Coef {
  float c;        // EMA coefficient
  float a16;      // a^16 (ratio across 16-step segments)
  float r16l;     // (a^16)^l for this lane's column index l
  float apix[8];  // a^(step+1) for this lane's 8 output steps (step = r + 8h)
  v2f   aF[4];    // A-matrix (L) fragments for the 4 chained K=4 WMMAs
};

__device__ __forceinline__ void make_coef(Coef& C, float c, int l, int h) {
  C.c = c;
  const float a  = 1.0f - c;
  const float a2 = a * a, a4 = a2 * a2, a8 = a4 * a4;
  C.a16 = a8 * a8;
  C.r16l = ipow_f(C.a16, l);
  float q = a * ipow_f(a, 8 * h);
#pragma unroll
  for (int r = 0; r < 8; ++r) { C.apix[r] = q; q *= a; }
  // A fragment layout (32-bit A 16x4, ISA 7.12.2): lane l<16 holds M=l,
  // VGPR0->K, VGPR1->K+1 with K = 4*cc + 2*h.
#pragma unroll
  for (int cc = 0; cc < 4; ++cc) {
    const int k0 = 4 * cc + 2 * h;
    const int d0 = l - k0, d1 = d0 - 1;
    v2f f;
    f.x = (d0 >= 0) ? ipow_f(a, d0) : 0.0f;
    f.y = (d1 >= 0) ? ipow_f(a, d1) : 0.0f;
    C.aF[cc] = f;
  }
}

// Scan one 256-float tile held in LDS. Returns this lane's 8 fixed-up outputs
// (segment l, steps r+8h) and advances the running carry (y just before tile).
__device__ __forceinline__ v8f tile_scan(const float* __restrict__ tile,
                                         const Coef& C, int l, int h,
                                         float& carry) {
  v8f D = {};
#pragma unroll
  for (int cc = 0; cc < 4; ++cc) {
    // B fragment (4x16): VGPR r holds row K = r (+2 for lanes 16-31), col = l.
    const int off = 16 * l + 4 * cc + 2 * h;
    v2f bF;
    bF.x = C.c * tile[off];
    bF.y = C.c * tile[off + 1];
    D = __builtin_amdgcn_wmma_f32_16x16x4_f32(
        /*neg_a=*/false, C.aF[cc], /*neg_b=*/false, bF,
        /*c_mod=*/(short)0, D, /*reuse_a=*/false, /*reuse_b=*/false);
  }
  // Column-end values: D[15][j] lives in VGPR7 of lane 16+j.
  const float colend = __shfl(D[7], 16 + l, 32);
  // Inclusive weighted scan S_l = sum_{k<=l} (a^16)^(l-k) * colend_k
  float S = colend, w = C.a16;
#pragma unroll
  for (int o = 1; o < 16; o <<= 1) {
    const float t = __shfl_up(S, o, 32);
    if (l >= o) S += w * t;
    w *= w;
  }
  const float Sprev = __shfl_up(S, 1, 32);
  const float P = C.r16l * carry + ((l > 0) ? Sprev : 0.0f);
  v8f Y;
#pragma unroll
  for (int r = 0; r < 8; ++r) Y[r] = D[r] + C.apix[r] * P;
  carry = __shfl(Y[7], 31, 32);  // y at (seg 15, step 15) = last tile sample
  return Y;
}

// Stage a 256-float tile into this wave's LDS slice (+ prefetch of next tile).
__device__ __forceinline__ void load_tile(const float* __restrict__ g,
                                          float* __restrict__ lds, int lane,
                                          bool prefetch_next) {
#if USE_ASYNC_LDS
  __builtin_amdgcn_global_load_async_to_lds_b128(
      (v4i*)(g + 4 * lane), (v4i*)(lds + 4 * lane), 0, 0);
  __builtin_amdgcn_global_load_async_to_lds_b128(
      (v4i*)(g + 128 + 4 * lane), (v4i*)(lds + 128 + 4 * lane), 0, 0);
#else
  const float4 a = ((const float4*)g)[lane];
  const float4 b = ((const float4*)g)[lane + 32];
  ((float4*)lds)[lane]      = a;
  ((float4*)lds)[lane + 32] = b;
#endif
  if (prefetch_next) __builtin_prefetch(g + 256 + 4 * lane, 0, 3);
}

__device__ __forceinline__ int widx(int coef, int arr, int row, int k) {
  return ((coef * 2 + arr) * NROWS + row) * NCHUNK + k;
}

// ---------------------------------------------------------------------------
// Pass 1: per-chunk EMA aggregates (zero-init scan value at chunk end).
// One wave per (signal, row, chunk); both coefficients from one data read.
// ---------------------------------------------------------------------------
__global__ void k_agg(const float* __restrict__ pred,
                      const float* __restrict__ targ,
                      float* __restrict__ agg, float cS, float cL) {
  const int wv = threadIdx.x >> 5, lane = threadIdx.x & 31;
  const int l = lane & 15, h = lane >> 4;
  const int w = blockIdx.x * 8 + wv;            // 0 .. 32767
  const int chunk = w & (NCHUNK - 1);
  const int row   = (w >> 10) & (NROWS - 1);
  const int arr   = (w >> 14) & 1;
  const float* x = (arr ? targ : pred) + (size_t)row * TLEN + (size_t)chunk * CHUNK;

  Coef CS, CL;
  make_coef(CS, cS, l, h);
  make_coef(CL, cL, l, h);

  __shared__ float lds[8 * 256];
  float* tile = lds + wv * 256;

  float carS = 0.0f, carL = 0.0f;
  for (int t = 0; t < 4; ++t) {
    wave_wait_ds();                       // prior tile reads done (WAR)
    load_tile(x + t * 256, tile, lane, t < 3);
    wave_wait_tile_published();           // tile visible to whole wave
    tile_scan(tile, CS, l, h, carS);
    tile_scan(tile, CL, l, h, carL);
  }
  if (lane == 0) {
    agg[widx(0, arr, row, chunk)] = carS;
    agg[widx(1, arr, row, chunk)] = carL;
  }
}

// ---------------------------------------------------------------------------
// Pass 2: serial scan over chunk aggregates -> exclusive per-chunk carries.
// 64 independent scans of 1024 steps, ratio a^1024.
// ---------------------------------------------------------------------------
__global__ void k_carry(const float* __restrict__ agg,
                        float* __restrict__ carry, float cS, float cL) {
  const int id = threadIdx.x;
  if (id >= 64) return;
  const int coef = id >> 5, arr = (id >> 4) & 1, row = id & 15;
  float a = 1.0f - (coef ? cL : cS);
  float A = a;
  for (int i = 0; i < 10; ++i) A *= A;          // a^1024
  const float* s = agg   + (size_t)widx(coef, arr, row, 0);
  float*       d = carry + (size_t)widx(coef, arr, row, 0);
  float c0 = 0.0f;
  for (int k = 0; k < NCHUNK; ++k) {
    d[k] = c0;                                  // exclusive: y before chunk k
    c0 = A * c0 + s[k];
  }
}

// ---------------------------------------------------------------------------
// Pass 3: per (row, chunk) wave: long-EMA logs for the shifted window
// (chunks kc+63 and kc+64, offset 535) staged in LDS, then short-EMA logs
// locally; accumulate |d - e| and emit one wave sum. Entirely wave-local.
// ---------------------------------------------------------------------------
__global__ void k_loss(const float* __restrict__ pred,
                       const float* __restrict__ targ,
                       const float* __restrict__ carry,
                       float* __restrict__ wsum, float cS, float cL) {
  const int wv = threadIdx.x >> 5, lane = threadIdx.x & 31;
  const int l = lane & 15, h = lane >> 4;
  const int w  = blockIdx.x * 8 + wv;           // 0 .. 16383
  const int kc = w & (NCHUNK - 1);
  const int row = (w >> 10) & (NROWS - 1);

  Coef CS, CL;
  make_coef(CS, cS, l, h);
  make_coef(CL, cL, l, h);

  __shared__ float tl[8 * 256];
  __shared__ float ew[8 * 1024];
  float* tile = tl + wv * 256;
  float* ewin = ew + wv * 1024;                 // e(t+SHIFT) for this chunk

  const int k0 = (kc + SHIFT_CH) & (NCHUNK - 1);
  const int k1 = (k0 + 1) & (NCHUNK - 1);

  // ---- long-EMA phase: pred writes log, targ subtracts log -> e in LDS ----
  for (int arr = 0; arr < 2; ++arr) {
    const float* base = (arr ? targ : pred) + (size_t)row * TLEN;
    for (int ci = 0; ci < 2; ++ci) {
      const int kk = ci ? k1 : k0;
      float car = carry[widx(1, arr, row, kk)];
      for (int t = 0; t < 4; ++t) {
        wave_wait_ds();
        load_tile(base + (size_t)kk * CHUNK + t * 256, tile, lane, t < 3);
        wave_wait_tile_published();
        v8f Y = tile_scan(tile, CL, l, h, car);
        const int ob = t * 256 + 16 * l + 8 * h; // offset within chunk kk
#pragma unroll
        for (int r = 0; r < 8; ++r) {
          const int wi = ci ? (ob + r + (CHUNK - SHIFT_OFF))  // +489
                            : (ob + r - SHIFT_OFF);           // -535
          if (wi >= 0 && wi < CHUNK) {
            const float lg = logf(Y[r]);
            if (arr == 0) ewin[wi] = lg; else ewin[wi] -= lg;
          }
        }
      }
    }
  }
  wave_wait_ds();   // ewin fully built for this wave

  // ---- short-EMA phase over this chunk; pair with e from LDS ----
  const float* bp = pred + (size_t)row * TLEN + (size_t)kc * CHUNK;
  const float* bt = targ + (size_t)row * TLEN + (size_t)kc * CHUNK;
  float carP = carry[widx(0, 0, row, kc)];
  float carT = carry[widx(0, 1, row, kc)];
  float acc = 0.0f;
  for (int t = 0; t < 4; ++t) {
    wave_wait_ds();
    load_tile(bp + t * 256, tile, lane, t < 3);
    wave_wait_tile_published();
    v8f Yp = tile_scan(tile, CS, l, h, carP);
    wave_wait_ds();
    load_tile(bt + t * 256, tile, lane, t < 3);
    wave_wait_tile_published();
    v8f Yt = tile_scan(tile, CS, l, h, carT);
    const int ob = t * 256 + 16 * l + 8 * h;
#pragma unroll
    for (int r = 0; r < 8; ++r) {
      const float d = logf(Yp[r]) - logf(Yt[r]);
      acc += fabsf(d - ewin[ob + r]);
    }
  }
#pragma unroll
  for (int o = 16; o > 0; o >>= 1) acc += __shfl_down(acc, o, 32);
  if (lane == 0) wsum[w] = acc;
}

// ---------------------------------------------------------------------------
// Pass 4: fixed-order reduction of the 16384 wave sums -> mean (deterministic).
// ---------------------------------------------------------------------------
__global__ void k_reduce(const float* __restrict__ wsum, float* __restrict__ out) {
  __shared__ float s[256];
  float a = 0.0f;
  for (int i = threadIdx.x; i < NROWS * NCHUNK; i += 256) a += wsum[i];
  s[threadIdx.x] = a;
  __syncthreads();
  for (int o = 128; o > 0; o >>= 1) {
    if (threadIdx.x < o) s[threadIdx.x] += s[threadIdx.x + o];
    __syncthreads();
  }
  if (threadIdx.x == 0) out[0] = s[0] * (1.0f / ((float)NROWS * (float)TLEN));
}

// ---------------------------------------------------------------------------
// Host side
// ---------------------------------------------------------------------------
extern "C" void kernel_launch(void* const* d_in, const int* in_sizes, int n_in,
                              void* d_out, int out_size, void* d_ws, size_t ws_size,
                              hipStream_t stream) {
  (void)in_sizes; (void)n_in; (void)out_size; (void)ws_size;
  const float* pred = (const float*)d_in[0];
  const float* targ = (const float*)d_in[1];
  float* out = (float*)d_out;

  // torchcomp-style coefficients, computed in double on the host.
  const double SRd = 44100.0;
  const float cS = (float)(1.0 - exp(-2200.0 / (50.0 * SRd)));
  const float cL = (float)(1.0 - exp(-2200.0 / (3000.0 * SRd)));

  // Workspace layout (floats): agg[65536] | carry[65536] | wsum[16384]
  float* agg   = (float*)d_ws;
  float* carry = agg + 4 * NROWS * NCHUNK;
  float* wsum  = carry + 4 * NROWS * NCHUNK;

  k_agg   <<<4096, 256, 0, stream>>>(pred, targ, agg, cS, cL);
  k_carry <<<1,     64, 0, stream>>>(agg, carry, cS, cL);
  k_loss  <<<2048, 256, 0, stream>>>(pred, targ, carry, wsum, cS, cL);
  k_reduce<<<1,    256, 0, stream>>>(wsum, out);
}